// CGABlock_38087769981516
// MI455X (gfx1250) — compile-verified
//
#include <hip/hip_runtime.h>
#include <math.h>

// MI455X / gfx1250, wave32. Fully fused CGA block: one workgroup per batch
// element, all intermediates in LDS, fp32 WMMA (16x16x4) for the GEMM-shaped
// contractions. Problem is latency/BW bound (~0.74 GFLOP, ~6.6 MB), so fp32
// accuracy is free.

typedef float v2f __attribute__((ext_vector_type(2)));
typedef float v8f __attribute__((ext_vector_type(8)));

#define BDIM 256
#define CIN 64
#define COUT 64
#define VN 25
#define MID 8
#define INTER 32
#define XS_STRIDE 32     // xs rows padded to 32 so WMMA N-tile reads stay in bounds
#define MIDROW 640       // 8 x 640 mid buffer (625 valid + pad)

__global__ __launch_bounds__(BDIM)
void cga_block_kernel(
    const float* __restrict__ x,      const float* __restrict__ A_static,
    const float* __restrict__ w1,     const float* __restrict__ b1,
    const float* __restrict__ w2,     const float* __restrict__ b2,
    const float* __restrict__ w3,     const float* __restrict__ b3,
    const float* __restrict__ diff_w, const float* __restrict__ diff_b,
    const float* __restrict__ edge_w, const float* __restrict__ edge_b,
    const float* __restrict__ att_w,  const float* __restrict__ att_b,
    const float* __restrict__ cc1_w,  const float* __restrict__ cc1_b,
    const float* __restrict__ bn_g,   const float* __restrict__ bn_b,
    const float* __restrict__ bn_m,   const float* __restrict__ bn_v,
    const float* __restrict__ cc2_w,  const float* __restrict__ cc2_b,
    const float* __restrict__ cs_w,   const float* __restrict__ cs_b,
    const float* __restrict__ alpha_p,
    float* __restrict__ out)
{
  __shared__ float xs[CIN * XS_STRIDE];   // input, padded cols
  __shared__ float As[VN * VN];           // A_static
  __shared__ float x1s[MID * VN];
  __shared__ float x2s[MID * VN];
  __shared__ float x3s[COUT * VN];
  __shared__ float S3[COUT];              // row sums of x3
  __shared__ float mid8[MID * MIDROW];    // shared: att-mid then A_dyn
  __shared__ float xatt[COUT * VN];
  __shared__ float xgcn[COUT * VN];       // pre c_att scaling
  __shared__ float xmS[COUT];
  __shared__ float hS[INTER];
  __shared__ float cattS[COUT];
  __shared__ float sattS[VN];

  const int bb   = blockIdx.x;
  const int t    = threadIdx.x;
  const int lane = t & 31;
  const int wv   = t >> 5;        // 8 waves
  const int half = lane >> 4;     // 0/1: which K/M half of the wave
  const int lcol = lane & 15;     // N column / M row within tile
  const int o0   = (wv >> 1) << 4;  // M tile base (4 tiles)
  const int n0   = (wv & 1) << 4;   // N tile base (2 tiles)
  const float alpha = alpha_p[0];

  if (t == 0) {
    __builtin_prefetch(w3, 0, 1);
    __builtin_prefetch(att_w, 0, 1);
    __builtin_prefetch(edge_w, 0, 1);
  }

  const float* xb = x + (size_t)bb * (CIN * VN);

  // ---- phase 1: stage x (zero-padded cols) and A_static into LDS ----
  for (int i = t; i < CIN * XS_STRIDE; i += BDIM) {
    int r = i >> 5, c = i & 31;
    xs[i] = (c < VN) ? xb[r * VN + c] : 0.0f;
  }
  for (int i = t; i < VN * VN; i += BDIM) As[i] = A_static[i];
  __syncthreads();

  // ---- phase 2: x1,x2 (scalar, 200 outs each) ; x3 via WMMA 64x64x32 ----
  for (int i = t; i < MID * VN; i += BDIM) {
    int m = i / VN, p = i % VN;
    float s1 = b1[m], s2 = b2[m];
    for (int c = 0; c < CIN; ++c) {
      float xv = xs[c * XS_STRIDE + p];
      s1 += w1[m * CIN + c] * xv;
      s2 += w2[m * CIN + c] * xv;
    }
    x1s[i] = s1; x2s[i] = s2;
  }
  {
    v8f acc = {};
    const int rowA = o0 + lcol;
    const int colB = n0 + lcol;
    for (int kk = 0; kk < CIN; kk += 4) {
      int ka = kk + 2 * half;
      v2f a, bm;
      a.x  = w3[rowA * CIN + ka];
      a.y  = w3[rowA * CIN + ka + 1];
      bm.x = xs[ka * XS_STRIDE + colB];
      bm.y = xs[(ka + 1) * XS_STRIDE + colB];
      acc = __builtin_amdgcn_wmma_f32_16x16x4_f32(false, a, false, bm,
                                                  (short)0, acc, false, false);
    }
    if (colB < VN) {
      for (int j = 0; j < 8; ++j) {
        int o = o0 + j + 8 * half;
        x3s[o * VN + colB] = acc[j] + b3[o];
      }
    }
  }
  __syncthreads();

  // ---- phase 3: S3 row sums; mid8 = tanh(x1_u * x2_v / 5) ----
  if (t < COUT) {
    float s = 0.f;
    for (int v = 0; v < VN; ++v) s += x3s[t * VN + v];
    S3[t] = s;
  }
  for (int i = t; i < MID * MIDROW; i += BDIM) {
    int m = i / MIDROW, p = i % MIDROW;
    float val = 0.f;
    if (p < VN * VN) {
      int u = p / VN, v = p % VN;
      val = tanhf(x1s[m * VN + u] * x2s[m * VN + v] * 0.2f);
    }
    mid8[i] = val;
  }
  __syncthreads();

  // ---- phase 4: attention path. Per wave: own (o-tile, v-tile), loop u ----
  {
    const int rowA = o0 + lcol;
    v2f a0, a1;
    a0.x = att_w[rowA * MID + 2 * half];     a0.y = att_w[rowA * MID + 2 * half + 1];
    a1.x = att_w[rowA * MID + 4 + 2 * half]; a1.y = att_w[rowA * MID + 5 + 2 * half];
    float xacc[8];
    for (int j = 0; j < 8; ++j) xacc[j] = 0.f;
    const int klo0 = 2 * half, klo1 = 4 + 2 * half;
    for (int u = 0; u < VN; ++u) {
      int p = u * VN + n0 + lcol;           // <= 631 < 640, pad is zero
      v2f bf0, bf1;
      bf0.x = mid8[klo0 * MIDROW + p]; bf0.y = mid8[(klo0 + 1) * MIDROW + p];
      bf1.x = mid8[klo1 * MIDROW + p]; bf1.y = mid8[(klo1 + 1) * MIDROW + p];
      v8f c = {};
      c = __builtin_amdgcn_wmma_f32_16x16x4_f32(false, a0, false, bf0, (short)0, c, false, false);
      c = __builtin_amdgcn_wmma_f32_16x16x4_f32(false, a1, false, bf1, (short)0, c, false, false);
      for (int j = 0; j < 8; ++j) {
        int o = o0 + j + 8 * half;
        xacc[j] += x3s[o * VN + u] * c[j];
      }
    }
    int v = n0 + lcol;
    if (v < VN) {
      for (int j = 0; j < 8; ++j) {
        int o = o0 + j + 8 * half;
        xatt[o * VN + v] = xacc[j] + att_b[o] * S3[o];
      }
    }
  }
  __syncthreads();

  // ---- phase 5: mid8 = A_dyn = tanh(dw0*(x1_u-x2_v) + dw1*(x2_u-x1_v) + db) ----
  for (int i = t; i < MID * MIDROW; i += BDIM) {
    int m = i / MIDROW, p = i % MIDROW;
    float val = 0.f;
    if (p < VN * VN) {
      int u = p / VN, v = p % VN;
      float d1 = x1s[m * VN + u] - x2s[m * VN + v];
      float d2 = x2s[m * VN + u] - x1s[m * VN + v];
      val = tanhf(diff_w[m * 2 + 0] * d1 + diff_w[m * 2 + 1] * d2 + diff_b[m]);
    }
    mid8[i] = val;
  }
  __syncthreads();

  // ---- phase 6: GCN path. Per wave: own (o-tile, u-tile), loop v ----
  {
    const int rowA = o0 + lcol;
    v2f a0, a1;
    a0.x = edge_w[rowA * MID + 2 * half];     a0.y = edge_w[rowA * MID + 2 * half + 1];
    a1.x = edge_w[rowA * MID + 4 + 2 * half]; a1.y = edge_w[rowA * MID + 5 + 2 * half];
    float gacc[8];
    for (int j = 0; j < 8; ++j) gacc[j] = 0.f;
    const int u_raw = n0 + lcol;
    const int ueff  = u_raw < VN ? u_raw : VN - 1;  // clamp pad columns
    const int klo0 = 2 * half, klo1 = 4 + 2 * half;
    for (int v = 0; v < VN; ++v) {
      int p = ueff * VN + v;
      v2f bf0, bf1;
      bf0.x = mid8[klo0 * MIDROW + p]; bf0.y = mid8[(klo0 + 1) * MIDROW + p];
      bf1.x = mid8[klo1 * MIDROW + p]; bf1.y = mid8[(klo1 + 1) * MIDROW + p];
      v8f c = {};
      c = __builtin_amdgcn_wmma_f32_16x16x4_f32(false, a0, false, bf0, (short)0, c, false, false);
      c = __builtin_amdgcn_wmma_f32_16x16x4_f32(false, a1, false, bf1, (short)0, c, false, false);
      for (int j = 0; j < 8; ++j) {
        int o = o0 + j + 8 * half;
        gacc[j] += x3s[o * VN + v] * c[j];
      }
    }
    if (u_raw < VN) {
      for (int j = 0; j < 8; ++j) {
        int o = o0 + j + 8 * half;
        float stat = 0.f;
        for (int v = 0; v < VN; ++v) stat += As[u_raw * VN + v] * x3s[o * VN + v];
        xgcn[o * VN + u_raw] = stat + alpha * (gacc[j] + edge_b[o] * S3[o]);
      }
    }
  }
  __syncthreads();

  // ---- phase 7: channel attention (tiny MLP on mean over v) ----
  if (t < COUT) {
    float s = 0.f;
    for (int v = 0; v < VN; ++v) s += xatt[t * VN + v];
    xmS[t] = s * (1.0f / VN);
  }
  __syncthreads();
  if (t < INTER) {
    float s = cc1_b[t];
    for (int o = 0; o < COUT; ++o) s += cc1_w[t * COUT + o] * xmS[o];
    s = (s - bn_m[t]) * (bn_g[t] * rsqrtf(bn_v[t] + 1e-5f)) + bn_b[t];
    hS[t] = 0.5f * s * (1.0f + erff(s * 0.70710678118654752f));  // exact GELU
  }
  __syncthreads();
  if (t < COUT) {
    float s = cc2_b[t];
    for (int i = 0; i < INTER; ++i) s += cc2_w[t * INTER + i] * hS[i];
    cattS[t] = 1.0f / (1.0f + expf(-s));
  }
  __syncthreads();

  // ---- phase 8: spatial attention on scaled x_gcn ----
  if (t < VN) {
    float s = cs_b[0];
    for (int o = 0; o < COUT; ++o) s += cs_w[o] * (xgcn[o * VN + t] * cattS[o]);
    sattS[t] = 1.0f / (1.0f + expf(-s));
  }
  __syncthreads();

  // ---- phase 9: out = x_gcn*c_att + x_att*s_att + x ----
  float* outb = out + (size_t)bb * (COUT * VN);
  for (int i = t; i < COUT * VN; i += BDIM) {
    int o = i / VN, v = i % VN;
    outb[i] = xgcn[i] * cattS[o] + xatt[i] * sattS[v] + xs[o * XS_STRIDE + v];
  }
}

extern "C" void kernel_launch(void* const* d_in, const int* in_sizes, int n_in,
                              void* d_out, int out_size, void* d_ws, size_t ws_size,
                              hipStream_t stream) {
  const float* x        = (const float*)d_in[0];
  const float* A_static = (const float*)d_in[1];
  const float* w1       = (const float*)d_in[2];
  const float* b1       = (const float*)d_in[3];
  const float* w2       = (const float*)d_in[4];
  const float* b2       = (const float*)d_in[5];
  const float* w3       = (const float*)d_in[6];
  const float* b3       = (const float*)d_in[7];
  const float* diff_w   = (const float*)d_in[8];
  const float* diff_b   = (const float*)d_in[9];
  const float* edge_w   = (const float*)d_in[10];
  const float* edge_b   = (const float*)d_in[11];
  const float* att_w    = (const float*)d_in[12];
  const float* att_b    = (const float*)d_in[13];
  const float* cc1_w    = (const float*)d_in[14];
  const float* cc1_b    = (const float*)d_in[15];
  const float* bn_g     = (const float*)d_in[16];
  const float* bn_b     = (const float*)d_in[17];
  const float* bn_m     = (const float*)d_in[18];
  const float* bn_v     = (const float*)d_in[19];
  const float* cc2_w    = (const float*)d_in[20];
  const float* cc2_b    = (const float*)d_in[21];
  const float* cs_w     = (const float*)d_in[22];
  const float* cs_b     = (const float*)d_in[23];
  const float* alpha_p  = (const float*)d_in[24];

  const int Bn = in_sizes[0] / (CIN * VN);  // 512
  cga_block_kernel<<<Bn, BDIM, 0, stream>>>(
      x, A_static, w1, b1, w2, b2, w3, b3, diff_w, diff_b,
      edge_w, edge_b, att_w, att_b, cc1_w, cc1_b,
      bn_g, bn_b, bn_m, bn_v, cc2_w, cc2_b, cs_w, cs_b, alpha_p,
      (float*)d_out);
}